// Codebook_34041910788865
// MI455X (gfx1250) — compile-verified
//
#include <hip/hip_runtime.h>

typedef __attribute__((ext_vector_type(2))) float v2f;
typedef __attribute__((ext_vector_type(8))) float v8f;
typedef int v4i __attribute__((ext_vector_type(4)));
typedef __attribute__((address_space(1))) v4i as1_v4i;   // global int4
typedef __attribute__((address_space(3))) v4i as3_v4i;   // LDS int4

#define DDIM 256
#define KCODES 1024
#define HWSZ 1024
#define NVEC 32768          // 32 * 32 * 32
#define NUMEL 8388608       // 32 * 256 * 32 * 32
#define EBROW 260           // 256 + 4 pad -> bank-conflict-free B-fragment reads
#define EBBUF (16 * EBROW)  // one 16-code slab (floats)

// ---------------------------------------------------------------------------
// Kernel 0: per-code squared norms  ||e_k||^2
// ---------------------------------------------------------------------------
__global__ __launch_bounds__(256) void enorm_kernel(const float* __restrict__ emb,
                                                    float* __restrict__ enorm) {
  int k = blockIdx.x * 256 + threadIdx.x;   // 4 blocks -> 1024 codes exactly
  const float4* p = (const float4*)(emb + (size_t)k * DDIM);
  float s = 0.f;
  #pragma unroll 4
  for (int i = 0; i < DDIM / 4; ++i) {
    float4 v = p[i];
    s += v.x * v.x + v.y * v.y + v.z * v.z + v.w * v.w;
  }
  enorm[k] = s;
}

// ---------------------------------------------------------------------------
// Kernel 1: fused GEMM (f32 WMMA) + argmin over 1024 codes.
// Block = 256 threads = 8 waves; each wave owns 16 rows (A resident in
// 128 VGPRs), block covers 128 rows; grid = 256 blocks.
// Embedding slabs (16 codes x 256 d, 16KB) are double-buffered in LDS and
// filled with async global->LDS b128 copies overlapped with the WMMAs.
// score[n][k] = ||e_k||^2 - 2 * z_n . e_k   (||z||^2 dropped: row-constant)
// ---------------------------------------------------------------------------
__global__ __launch_bounds__(256) void vq_argmin_kernel(
    const float* __restrict__ z, const float* __restrict__ emb,
    const float* __restrict__ enorm, int* __restrict__ idx_out,
    float* __restrict__ idxf_out) {
  __shared__ float Eb[2 * EBBUF];   // 2 x 16.6KB, row-major [code][260]

  const int tid  = threadIdx.x;
  const int w    = tid >> 5;
  const int l    = tid & 31;
  const int half = l >> 4;       // 0: lanes 0-15, 1: lanes 16-31
  const int l15  = l & 15;

  const int row_base = blockIdx.x * 128 + w * 16;  // multiple of 16; HW=1024 -> same b
  const int b   = row_base >> 10;
  const int hw0 = row_base & 1023;
  const float* zb = z + (size_t)b * (DDIM * HWSZ) + hw0;

  // ---- A tile in registers: WMMA f32 16x16x4 A-fragment layout.
  // lane l<16 holds row l dims {4s, 4s+1}; lane l+16 holds dims {4s+2, 4s+3}.
  v2f areg[64];
  {
    const float* zl = zb + l15 + (size_t)(half * 2) * HWSZ;
    #pragma unroll
    for (int s = 0; s < 64; ++s) {
      areg[s].x = zl[(size_t)s * (4 * HWSZ)];
      areg[s].y = zl[(size_t)s * (4 * HWSZ) + HWSZ];
    }
  }

  // per-thread async-staging offsets: thread copies 4 x 16B of the 16KB slab
  int lofs[4], gq[4];
  #pragma unroll
  for (int j = 0; j < 4; ++j) {
    int q = tid + 256 * j;                      // float4 index within slab
    lofs[j] = (q >> 6) * EBROW + (q & 63) * 4;  // [code][d4] padded row-major
    gq[j]   = q * 4;                            // float offset within slab
  }

  // prologue: async-fill buffer 0 with chunk 0 (slab is contiguous in emb)
  #pragma unroll
  for (int j = 0; j < 4; ++j)
    __builtin_amdgcn_global_load_async_to_lds_b128(
        (as1_v4i*)(emb + gq[j]), (as3_v4i*)(Eb + lofs[j]), 0, 0);

  float minv[8];
  int   mini[8];
  #pragma unroll
  for (int v = 0; v < 8; ++v) { minv[v] = 3.4028235e38f; mini[v] = 0; }

  for (int chunk = 0; chunk < 64; ++chunk) {
    asm volatile("s_wait_asynccnt 0x0" ::: "memory");  // my copies landed
    __syncthreads();                                   // all copies landed; prev buffer free

    if (chunk + 1 < 64) {     // overlap next slab's copy with this chunk's WMMAs
      const float* gp = emb + (size_t)(chunk + 1) * (16 * DDIM);
      float* lb = Eb + ((chunk + 1) & 1) * EBBUF;
      #pragma unroll
      for (int j = 0; j < 4; ++j)
        __builtin_amdgcn_global_load_async_to_lds_b128(
            (as1_v4i*)(gp + gq[j]), (as3_v4i*)(lb + lofs[j]), 0, 0);
    }

    // B fragment: code = l15, dims {4s + 2*half, +1} -> adjacent pair, one ds_load_b64
    const float* Bp = Eb + (chunk & 1) * EBBUF + l15 * EBROW + half * 2;
    v8f acc = {};
    #pragma unroll
    for (int s = 0; s < 64; ++s) {          // full D=256 reduction, K=4 per WMMA
      v2f bf = *(const v2f*)(Bp + 4 * s);
      acc = __builtin_amdgcn_wmma_f32_16x16x4_f32(
          false, areg[s], false, bf, (short)0, acc, false, false);
    }

    int code = chunk * 16 + l15;
    float en = enorm[code];
    #pragma unroll
    for (int v = 0; v < 8; ++v) {
      float s = en - 2.0f * acc[v];
      if (s < minv[v]) { minv[v] = s; mini[v] = code; }  // codes ascend -> first-min kept
    }
  }

  // cross-lane argmin over the 16 code-slots sharing each row (lanes 0-15 / 16-31)
  #pragma unroll
  for (int v = 0; v < 8; ++v) {
    float mv = minv[v];
    int   mi = mini[v];
    #pragma unroll
    for (int m = 1; m <= 8; m <<= 1) {
      float ov = __shfl_xor(mv, m, 32);
      int   oi = __shfl_xor(mi, m, 32);
      if (ov < mv || (ov == mv && oi < mi)) { mv = ov; mi = oi; }
    }
    if (l15 == 0) {
      int row = row_base + v + half * 8;   // C layout: VGPR v -> M = v + 8*half
      idx_out[row]  = mi;
      idxf_out[row] = (float)mi;
    }
  }
}

// ---------------------------------------------------------------------------
// Kernel 2: gather z_q (NCHW, straight-through arithmetic) + loss partials
// Block = 256 threads handles 64 consecutive n; grid = 512 blocks.
// ---------------------------------------------------------------------------
__global__ __launch_bounds__(256) void gather_kernel(
    const float* __restrict__ z, const float* __restrict__ emb,
    const int* __restrict__ idx, float* __restrict__ zq_out,
    float* __restrict__ partials) {
  __shared__ int   kk[64];
  __shared__ float red[256];
  const int tid = threadIdx.x;
  const int n0  = blockIdx.x * 64;     // 64 | 1024 -> same b across block
  const int b   = n0 >> 10;
  const int hw0 = n0 & 1023;
  if (tid < 64) kk[tid] = idx[n0 + tid];
  __syncthreads();

  const int hwo   = tid & 63;
  const int cbase = tid >> 6;
  const float* erow = emb + (size_t)kk[hwo] * DDIM;
  const size_t ebase = ((size_t)b * DDIM) << 10;

  float acc = 0.f;
  for (int co = 0; co < 64; ++co) {
    int c = co * 4 + cbase;
    float val = erow[c];
    size_t e = ebase + ((size_t)c << 10) + hw0 + hwo;  // NCHW flat index
    float zf = z[e];
    zq_out[e] = zf + (val - zf);        // straight-through: z + sg(z_q - z)
    float diff = val - zf;              // loss uses raw z_q
    acc += diff * diff;
  }

  red[tid] = acc;
  __syncthreads();
  for (int s = 128; s > 0; s >>= 1) {
    if (tid < s) red[tid] += red[tid + s];
    __syncthreads();
  }
  if (tid == 0) partials[blockIdx.x] = red[0];
}

// ---------------------------------------------------------------------------
// Kernel 3: deterministic final loss reduction (512 partials)
// ---------------------------------------------------------------------------
__global__ __launch_bounds__(256) void loss_kernel(const float* __restrict__ partials,
                                                   float* __restrict__ loss_out) {
  __shared__ float red[256];
  const int tid = threadIdx.x;
  red[tid] = partials[tid * 2] + partials[tid * 2 + 1];
  __syncthreads();
  for (int s = 128; s > 0; s >>= 1) {
    if (tid < s) red[tid] += red[tid + s];
    __syncthreads();
  }
  if (tid == 0) loss_out[0] = red[0] * (1.25f / (float)NUMEL);
}

// ---------------------------------------------------------------------------
extern "C" void kernel_launch(void* const* d_in, const int* in_sizes, int n_in,
                              void* d_out, int out_size, void* d_ws, size_t ws_size,
                              hipStream_t stream) {
  const float* z   = (const float*)d_in[0];   // [32,256,32,32] f32
  const float* emb = (const float*)d_in[1];   // [1024,256] f32

  float* out   = (float*)d_out;
  float* zq    = out;                 // 8388608 floats, NCHW
  float* idxf  = out + NUMEL;         // 32768 indices (as float)
  float* loss  = out + NUMEL + NVEC;  // 1 float

  char* ws = (char*)d_ws;
  float* enorm    = (float*)ws;                      // 1024 f32
  int*   idx      = (int*)(ws + 4096);               // 32768 i32
  float* partials = (float*)(ws + 4096 + NVEC * 4);  // 512 f32

  enorm_kernel<<<4, 256, 0, stream>>>(emb, enorm);
  vq_argmin_kernel<<<NVEC / 128, 256, 0, stream>>>(z, emb, enorm, idx, idxf);
  gather_kernel<<<NVEC / 64, 256, 0, stream>>>(z, emb, idx, zq, partials);
  loss_kernel<<<1, 256, 0, stream>>>(partials, loss);
}